// HolyLossNetwork_70712341561808
// MI455X (gfx1250) — compile-verified
//
#include <hip/hip_runtime.h>
#include <hip/hip_bf16.h>
#include <math.h>

typedef float v2f __attribute__((ext_vector_type(2)));
typedef float v8f __attribute__((ext_vector_type(8)));

#define NEP      256
#define NPER     32
#define DIN      512
#define DF       512
#define TDIM     1026      // 2*(DF+1)
#define ZSTRIDE  1040      // TDIM padded to multiple of 16
#define XK       (4*ZSTRIDE) // 4160: concatenated feature length for score GEMM
#define LOG_2PIE 2.8378770664093453f

__device__ __forceinline__ v8f wmma_f32(v2f a, v2f b, v8f c) {
    // V_WMMA_F32_16X16X4_F32 : D = A(16x4 f32) x B(4x16 f32) + C(16x16 f32)
    return __builtin_amdgcn_wmma_f32_16x16x4_f32(
        /*neg_a=*/false, a, /*neg_b=*/false, b,
        /*c_mod=*/(short)0, c, /*reuse_a=*/false, /*reuse_b=*/false);
}

// ---------------------------------------------------------------------------
// Kernel 1: per-episode  phi = relu(inputs[e] @ W_fe + b_fe)  (32x512),
// reduced immediately to column mean/std over the 32 rows; z row written
// (stride ZSTRIDE, zero-padded).  One workgroup = one (set, episode).
// 8 waves; wave w owns 64 output columns (4 N-tiles) x 2 M-tiles.
// A (32 x 256 K-chunk) staged in LDS, stride 260 (bank-conflict-free).
// ---------------------------------------------------------------------------
__global__ __launch_bounds__(256) void feat_stats_kernel(
    const float* __restrict__ in_tr, const float* __restrict__ tg_tr,
    const float* __restrict__ in_te, const float* __restrict__ tg_te,
    const float* __restrict__ W_fe,  const float* __restrict__ b_fe,
    float* __restrict__ z_ws)
{
    __shared__ float sA[32 * 260];            // 33,280 bytes
    const int e   = blockIdx.x;
    const int set = blockIdx.y;
    const float* inp = set ? in_te : in_tr;
    const float* tgt = set ? tg_te : tg_tr;
    float* zrow = z_ws + ((size_t)set * NEP + e) * ZSTRIDE;

    const int tid  = threadIdx.x;
    const int wave = tid >> 5;
    const int lane = tid & 31;
    const int lm   = lane & 15;
    const int h    = lane >> 4;
    const int nb   = wave * 64;

    const float* Ae = inp + (size_t)e * NPER * DIN;

    v8f acc[2][4] = {};

    for (int kc = 0; kc < DIN; kc += 256) {
        // stage A chunk (32 x 256) into LDS
        for (int idx = tid; idx < 32 * 256; idx += 256) {
            int r = idx >> 8, c = idx & 255;
            sA[r * 260 + c] = Ae[(size_t)r * DIN + kc + c];
        }
        __syncthreads();

        for (int k0 = 0; k0 < 256; k0 += 4) {
            const int ka = k0 + 2 * h;           // lane-half K offset
            v2f a0 = *(const v2f*)&sA[lm * 260 + ka];
            v2f a1 = *(const v2f*)&sA[(16 + lm) * 260 + ka];
            const int kg = kc + ka;
#pragma unroll
            for (int j = 0; j < 4; ++j) {
                const int n = nb + j * 16 + lm;
                v2f b;
                b.x = W_fe[(size_t)kg       * DF + n];
                b.y = W_fe[(size_t)(kg + 1) * DF + n];
                acc[0][j] = wmma_f32(a0, b, acc[0][j]);
                acc[1][j] = wmma_f32(a1, b, acc[1][j]);
            }
        }
        __syncthreads();
    }

    // relu + bias, then column mean / std (ddof=1) over the 32 rows
#pragma unroll
    for (int j = 0; j < 4; ++j) {
        const int n = nb + j * 16 + lm;
        const float bias = b_fe[n];
        float s = 0.f, q = 0.f;
#pragma unroll
        for (int mt = 0; mt < 2; ++mt) {
#pragma unroll
            for (int r = 0; r < 8; ++r) {
                float v = acc[mt][j][r] + bias;
                v = fmaxf(v, 0.f);
                s += v;
                q += v * v;
            }
        }
        s += __shfl_xor(s, 16, 32);     // combine the two lane-halves (other 16 rows)
        q += __shfl_xor(q, 16, 32);
        const float mean = s * (1.0f / 32.0f);
        const float var  = (q - 32.0f * mean * mean) * (1.0f / 31.0f);
        const float sd   = sqrtf(fmaxf(var, 0.0f));
        if (h == 0) {
            zrow[1 + n]   = mean;       // x = [target, phi_0..phi_511]
            zrow[514 + n] = sd;         // std block starts at 513; phi_n -> 514+n
        }
    }

    // targets column (x column 0)
    if (tid == 0) {
        float s = 0.f, q = 0.f;
        for (int i = 0; i < NPER; ++i) {
            float v = tgt[(size_t)e * NPER + i];
            s += v; q += v * v;
        }
        const float mean = s * (1.0f / 32.0f);
        const float var  = (q - 32.0f * mean * mean) * (1.0f / 31.0f);
        zrow[0]   = mean;
        zrow[513] = sqrtf(fmaxf(var, 0.0f));
    }
    // zero the padding columns so GEMM K-tails need no guards
    if (tid < ZSTRIDE - TDIM) zrow[TDIM + tid] = 0.0f;
}

// ---------------------------------------------------------------------------
// Kernel 2: lv = clip(z @ W_lv + b_lv, -9, -2).  M=256, N=1026, K=1026.
// One wave per 16x16 output tile; K padded to ZSTRIDE (z padding is zero).
// ---------------------------------------------------------------------------
__global__ __launch_bounds__(32) void logvar_kernel(
    const float* __restrict__ z_ws, const float* __restrict__ W_lv,
    const float* __restrict__ b_lv, float* __restrict__ lv_ws)
{
    const int nt  = blockIdx.x;   // 0..64
    const int mt  = blockIdx.y;   // 0..15
    const int set = blockIdx.z;
    const float* Z  = z_ws  + (size_t)set * NEP * ZSTRIDE;
    float*       LV = lv_ws + (size_t)set * NEP * ZSTRIDE;

    const int lane = threadIdx.x;
    const int lm   = lane & 15;
    const int h    = lane >> 4;
    const int m    = mt * 16 + lm;     // A row
    const int n    = nt * 16 + lm;     // B/out column

    v8f acc = {};
    for (int k0 = 0; k0 < ZSTRIDE; k0 += 4) {
        const int ka = k0 + 2 * h;
        v2f a = *(const v2f*)&Z[(size_t)m * ZSTRIDE + ka];
        v2f b;
        b.x = (ka     < TDIM && n < TDIM) ? W_lv[(size_t)ka       * TDIM + n] : 0.0f;
        b.y = (ka + 1 < TDIM && n < TDIM) ? W_lv[(size_t)(ka + 1) * TDIM + n] : 0.0f;
        acc = wmma_f32(a, b, acc);
    }

    if (n < TDIM) {
        const float bias = b_lv[n];
#pragma unroll
        for (int r = 0; r < 8; ++r) {
            const int row = mt * 16 + r + 8 * h;     // C/D layout
            float v = acc[r] + bias;
            v = fminf(fmaxf(v, -9.0f), -2.0f);
            LV[(size_t)row * ZSTRIDE + n] = v;
        }
    }
}

// ---------------------------------------------------------------------------
// Kernel 3: build score-GEMM operand rows + per-episode constants.
//   train: X = [P, F, -2*mu, -2*G],  rowc = 0.5*Q + H - T
//   test : Y = [F, P,  G,     mu ],  colc = 0.5*Q + H
// with P = e^lv + mu^2, F = e^-lv, G = mu*e^-lv, Q = sum mu^2 e^-lv,
// H = sum lv + 0.5*T*log(2*pi*e).  Padding columns zeroed.
// ---------------------------------------------------------------------------
__global__ __launch_bounds__(256) void prep_kernel(
    const float* __restrict__ z_ws, const float* __restrict__ lv_ws,
    float* __restrict__ X, float* __restrict__ Y,
    float* __restrict__ rowc, float* __restrict__ colc)
{
    const int e   = blockIdx.x;
    const int set = blockIdx.y;
    const float* zr = z_ws  + ((size_t)set * NEP + e) * ZSTRIDE;
    const float* lr = lv_ws + ((size_t)set * NEP + e) * ZSTRIDE;
    float* outr = (set ? Y : X) + (size_t)e * XK;

    __shared__ float sq[256], sh[256];
    float q = 0.f, hs = 0.f;

    for (int t = threadIdx.x; t < ZSTRIDE; t += 256) {
        float s0 = 0.f, s1 = 0.f, s2 = 0.f, s3 = 0.f;
        if (t < TDIM) {
            const float mu  = zr[t];
            const float l   = lr[t];
            const float el  = expf(l);
            const float eml = expf(-l);
            const float P   = el + mu * mu;
            const float G   = mu * eml;
            if (set == 0) { s0 = P;   s1 = eml; s2 = -2.0f * mu; s3 = -2.0f * G; }
            else          { s0 = eml; s1 = P;   s2 = G;          s3 = mu;        }
            q  += mu * mu * eml;
            hs += l;
        }
        outr[t]               = s0;
        outr[ZSTRIDE + t]     = s1;
        outr[2 * ZSTRIDE + t] = s2;
        outr[3 * ZSTRIDE + t] = s3;
    }

    sq[threadIdx.x] = q;
    sh[threadIdx.x] = hs;
    __syncthreads();
    for (int s = 128; s > 0; s >>= 1) {
        if (threadIdx.x < s) {
            sq[threadIdx.x] += sq[threadIdx.x + s];
            sh[threadIdx.x] += sh[threadIdx.x + s];
        }
        __syncthreads();
    }
    if (threadIdx.x == 0) {
        const float H = sh[0] + 0.5f * (float)TDIM * LOG_2PIE;
        if (set == 0) rowc[e] = 0.5f * sq[0] + H - (float)TDIM;
        else          colc[e] = 0.5f * sq[0] + H;
    }
}

// ---------------------------------------------------------------------------
// Kernel 4: score = 0.5 * (X @ Y^T) + rowc[i] + colc[j].  256x256xXK WMMA.
// ---------------------------------------------------------------------------
__global__ __launch_bounds__(32) void score_kernel(
    const float* __restrict__ X, const float* __restrict__ Y,
    const float* __restrict__ rowc, const float* __restrict__ colc,
    float* __restrict__ score)
{
    const int nt = blockIdx.x;
    const int mt = blockIdx.y;
    const int lane = threadIdx.x;
    const int lm   = lane & 15;
    const int h    = lane >> 4;
    const int m    = mt * 16 + lm;
    const int n    = nt * 16 + lm;

    v8f acc = {};
    for (int k0 = 0; k0 < XK; k0 += 4) {
        const int ka = k0 + 2 * h;
        v2f a = *(const v2f*)&X[(size_t)m * XK + ka];
        v2f b = *(const v2f*)&Y[(size_t)n * XK + ka];   // B = Y^T
        acc = wmma_f32(a, b, acc);
    }

#pragma unroll
    for (int r = 0; r < 8; ++r) {
        const int row = mt * 16 + r + 8 * h;
        score[(size_t)row * NEP + n] = 0.5f * acc[r] + rowc[row] + colc[n];
    }
}

// ---------------------------------------------------------------------------
// Kernel 5: probs = log_softmax(-score, axis=1); classes = arange(256)
// ---------------------------------------------------------------------------
__global__ __launch_bounds__(256) void softmax_kernel(
    const float* __restrict__ score, float* __restrict__ probs,
    int* __restrict__ classes)
{
    const int i = blockIdx.x;
    const int j = threadIdx.x;
    __shared__ float red[256];

    const float x = -score[(size_t)i * NEP + j];
    red[j] = x;
    __syncthreads();
    for (int s = 128; s > 0; s >>= 1) {
        if (j < s) red[j] = fmaxf(red[j], red[j + s]);
        __syncthreads();
    }
    const float mx = red[0];
    __syncthreads();
    red[j] = expf(x - mx);
    __syncthreads();
    for (int s = 128; s > 0; s >>= 1) {
        if (j < s) red[j] += red[j + s];
        __syncthreads();
    }
    const float lse = mx + logf(red[0]);
    probs[(size_t)i * NEP + j] = x - lse;
    if (j == 0) classes[i] = i;
}

// ---------------------------------------------------------------------------
extern "C" void kernel_launch(void* const* d_in, const int* in_sizes, int n_in,
                              void* d_out, int out_size, void* d_ws, size_t ws_size,
                              hipStream_t stream)
{
    const float* tr_in = (const float*)d_in[0];
    const float* tr_tg = (const float*)d_in[1];
    const float* te_in = (const float*)d_in[2];
    const float* te_tg = (const float*)d_in[3];
    const float* W_fe  = (const float*)d_in[4];
    const float* b_fe  = (const float*)d_in[5];
    const float* W_lv  = (const float*)d_in[6];
    const float* b_lv  = (const float*)d_in[7];

    float* ws    = (float*)d_ws;
    float* z     = ws;                              // 2 * 256 * 1040
    float* lv    = z  + 2 * NEP * ZSTRIDE;          // 2 * 256 * 1040
    float* X     = lv + 2 * NEP * ZSTRIDE;          // 256 * 4160
    float* Y     = X  + (size_t)NEP * XK;           // 256 * 4160
    float* rowc  = Y  + (size_t)NEP * XK;           // 256
    float* colc  = rowc + NEP;                      // 256
    float* score = colc + NEP;                      // 256 * 256

    feat_stats_kernel<<<dim3(NEP, 2), 256, 0, stream>>>(
        tr_in, tr_tg, te_in, te_tg, W_fe, b_fe, z);

    logvar_kernel<<<dim3(65, 16, 2), 32, 0, stream>>>(z, W_lv, b_lv, lv);

    prep_kernel<<<dim3(NEP, 2), 256, 0, stream>>>(z, lv, X, Y, rowc, colc);

    score_kernel<<<dim3(16, 16), 32, 0, stream>>>(X, Y, rowc, colc, score);

    softmax_kernel<<<NEP, 256, 0, stream>>>(
        score, (float*)d_out, ((int*)d_out) + NEP * NEP);
}